// GCNDecoder_5789615915455
// MI455X (gfx1250) — compile-verified
//
#include <hip/hip_runtime.h>
#include <hip/hip_bf16.h>

typedef __attribute__((ext_vector_type(2))) float v2f;
typedef __attribute__((ext_vector_type(8))) float v8f;

#define DHID   128
#define NPM    32          // nodes per molecule
#define MZ     10          // output classes
#define XP     132         // padded pitch for 128-wide rows (avoid bank aliasing)
#define SP     36          // padded pitch for 32-wide rows
#define EPSF   1e-7f

__device__ __forceinline__ v8f wmma_f32(v2f a, v2f b, v8f c) {
  // D = A(16x4,f32) * B(4x16,f32) + C(16x16,f32)
  return __builtin_amdgcn_wmma_f32_16x16x4_f32(false, a, false, b, (short)0, c,
                                               false, false);
}

// Async global->LDS B128 copy (ASYNCcnt-tracked DMA, no VGPR round-trip).
// Generic pointers to __shared__ carry the LDS byte offset in bits [31:0]
// (ISA 10.2), so truncation yields the VDST operand.
__device__ __forceinline__ void async_copy_b128(void* lds_dst, const void* gsrc) {
  unsigned lds_off = (unsigned)(unsigned long long)lds_dst;
  asm volatile("global_load_async_to_lds_b128 %0, %1, off"
               :: "v"(lds_off), "v"(gsrc) : "memory");
}
__device__ __forceinline__ void async_fence() {
  asm volatile("s_wait_asynccnt 0x0" ::: "memory");
}

__global__ __launch_bounds__(128)
void gcn_decoder_fused(const float* __restrict__ h,
                       const float* __restrict__ node_mask,
                       const float* __restrict__ edge_mask,
                       const float* __restrict__ W0, const float* __restrict__ b0,
                       const float* __restrict__ W1, const float* __restrict__ b1,
                       const float* __restrict__ Wout, const float* __restrict__ bout,
                       float* __restrict__ out)
{
  __shared__ float sX  [NPM * XP];   // node features, row-major [node][dim]
  __shared__ float sWt [DHID * XP];  // W transposed [n][k] -> contiguous B frags
  __shared__ float sSt [DHID * SP];  // support transposed [dim][node]
  __shared__ float sM  [NPM * SP];   // 32x32 edge-mask block [i][j]
  __shared__ float sWoT[16 * XP];    // Wout^T, N padded 10->16
  __shared__ float sNM [NPM];
  __shared__ float sF  [NPM];        // logmap0 per-node factor

  const int mol      = blockIdx.x;
  const int t        = threadIdx.x;
  const int lane     = t & 31;
  const int wave     = t >> 5;
  const int nodeBase = mol * NPM;

  // ---------------- stage molecule-local data into LDS ----------------
  {
    const float4* src = (const float4*)(h + (size_t)nodeBase * DHID);
#pragma unroll
    for (int j = 0; j < 8; ++j) {                 // 1024 float4 = 32x128
      int f = t + j * 128;
      int node = f >> 5, dq = f & 31;
      async_copy_b128(&sX[node * XP + dq * 4], src + f);
    }
  }
  {
    const float4* src = (const float4*)(edge_mask + (size_t)mol * NPM * NPM);
#pragma unroll
    for (int j = 0; j < 2; ++j) {                 // 256 float4 = 32x32
      int f = t + j * 128;
      int r = f >> 3, cq = f & 7;
      async_copy_b128(&sM[r * SP + cq * 4], src + f);
    }
  }
  if (t < NPM) sNM[t] = node_mask[nodeBase + t];
  for (int idx = t; idx < 16 * DHID; idx += 128) { // Wout^T with zero pad
    int z = idx & 15, k = idx >> 4;
    sWoT[z * XP + k] = (z < MZ) ? Wout[k * MZ + z] : 0.0f;
  }
  async_fence();   // sX / sM resident before first GEMM

  // tile ownership: wave -> one 16-row half (mt), four 16-col tiles (nt0..nt0+3)
  const int mt   = wave >> 1;
  const int nt0  = (wave & 1) * 4;
  const int half = lane >> 4;      // selects K pair / M offset 8
  const int lr   = lane & 15;      // row (A) / col (B,C)

  const float* Wl[2] = {W0, W1};
  const float* bl[2] = {b0, b1};

  for (int layer = 0; layer < 2; ++layer) {
    __syncthreads();
    {  // stage W^T : global row-major [k][n] -> LDS [n][k]
      const float4* src = (const float4*)Wl[layer];
#pragma unroll 4
      for (int j = 0; j < 32; ++j) {              // 4096 float4 = 128x128
        int f = t + j * 128;
        int k = f >> 5, n0 = (f & 31) * 4;
        float4 v = src[f];
        sWt[(n0 + 0) * XP + k] = v.x;
        sWt[(n0 + 1) * XP + k] = v.y;
        sWt[(n0 + 2) * XP + k] = v.z;
        sWt[(n0 + 3) * XP + k] = v.w;
      }
    }
    __syncthreads();

    // -------- support = X @ W + b  -> sSt (dim-major) --------
    {
      // hoist biases: issue global loads before the long WMMA loop so the
      // epilogue never waits on loadcnt+dscnt together
      float bias[4];
#pragma unroll
      for (int i = 0; i < 4; ++i) bias[i] = bl[layer][(nt0 + i) * 16 + lr];

      v8f acc[4] = {{0,0,0,0,0,0,0,0},{0,0,0,0,0,0,0,0},
                    {0,0,0,0,0,0,0,0},{0,0,0,0,0,0,0,0}};
      const float* arow = &sX[(mt * 16 + lr) * XP + 2 * half];
#pragma unroll 4
      for (int k0 = 0; k0 < DHID; k0 += 4) {
        v2f a = *(const v2f*)(arow + k0);
#pragma unroll
        for (int i = 0; i < 4; ++i) {
          v2f b = *(const v2f*)&sWt[((nt0 + i) * 16 + lr) * XP + k0 + 2 * half];
          acc[i] = wmma_f32(a, b, acc[i]);
        }
      }
#pragma unroll
      for (int i = 0; i < 4; ++i) {
        int n = (nt0 + i) * 16 + lr;
#pragma unroll
        for (int r = 0; r < 8; ++r) {
          int node = mt * 16 + r + 8 * half;
          sSt[n * SP + node] = acc[i][r] + bias[i];
        }
      }
    }
    __syncthreads();

    // -------- agg = M @ support ; relu * node_mask -> sX --------
    {
      v8f acc[4] = {{0,0,0,0,0,0,0,0},{0,0,0,0,0,0,0,0},
                    {0,0,0,0,0,0,0,0},{0,0,0,0,0,0,0,0}};
      const float* arow = &sM[(mt * 16 + lr) * SP + 2 * half];
#pragma unroll
      for (int k0 = 0; k0 < NPM; k0 += 4) {
        v2f a = *(const v2f*)(arow + k0);
#pragma unroll
        for (int i = 0; i < 4; ++i) {
          v2f b = *(const v2f*)&sSt[((nt0 + i) * 16 + lr) * SP + k0 + 2 * half];
          acc[i] = wmma_f32(a, b, acc[i]);
        }
      }
#pragma unroll
      for (int i = 0; i < 4; ++i) {
        int n = (nt0 + i) * 16 + lr;
#pragma unroll
        for (int r = 0; r < 8; ++r) {
          int node = mt * 16 + r + 8 * half;
          float v = fmaxf(acc[i][r], 0.0f) * sNM[node];
          sX[node * XP + n] = v;
        }
      }
    }
  }
  __syncthreads();

  // -------- logmap0 factors: f = arctanh(min(max(|x|,eps),1-eps)) / max(|x|,eps)
  {
    int node = t >> 2, q = t & 3;
    const float* xr = &sX[node * XP + q * 32];
    float s = 0.0f;
#pragma unroll
    for (int i = 0; i < 32; i += 2) {
      v2f v = *(const v2f*)(xr + i);
      s += v.x * v.x + v.y * v.y;
    }
    s += __shfl_xor(s, 1, 32);
    s += __shfl_xor(s, 2, 32);
    if (q == 0) {
      float nc = fmaxf(sqrtf(s), EPSF);
      float tt = fminf(nc, 1.0f - EPSF);
      sF[node] = atanhf(tt) / nc;
    }
  }
  __syncthreads();

  // -------- out = (f .* X) @ Wout^T + bout  (2 tiles -> waves 0,1) --------
  if (wave < 2) {
    const int z = lr;
    const float bo = bout[z < MZ ? z : (MZ - 1)];   // unpredicated, EXEC stays full
    v8f acc = {0,0,0,0,0,0,0,0};
    const int mrow = wave * 16 + lr;
    const float fac = sF[mrow];
    const float* arow = &sX[mrow * XP + 2 * half];
    const float* brow = &sWoT[lr * XP + 2 * half];
#pragma unroll 4
    for (int k0 = 0; k0 < DHID; k0 += 4) {
      v2f a = *(const v2f*)(arow + k0);
      a.x *= fac; a.y *= fac;
      v2f b = *(const v2f*)(brow + k0);
      acc = wmma_f32(a, b, acc);
    }
    if (z < MZ) {
#pragma unroll
      for (int r = 0; r < 8; ++r) {
        int node = wave * 16 + r + 8 * half;
        out[(size_t)(nodeBase + node) * MZ + z] = acc[r] + bo;
      }
    }
  }
}

extern "C" void kernel_launch(void* const* d_in, const int* in_sizes, int n_in,
                              void* d_out, int out_size, void* d_ws, size_t ws_size,
                              hipStream_t stream) {
  // input order: h, distances(unused), edges(fixed fully-connected structure,
  // folded into the per-molecule 32x32 mask GEMM), node_mask, edge_mask,
  // W0, b0, W1, b1, Wout, bout
  const float* h         = (const float*)d_in[0];
  const float* node_mask = (const float*)d_in[3];
  const float* edge_mask = (const float*)d_in[4];
  const float* W0        = (const float*)d_in[5];
  const float* b0        = (const float*)d_in[6];
  const float* W1        = (const float*)d_in[7];
  const float* b1        = (const float*)d_in[8];
  const float* Wout      = (const float*)d_in[9];
  const float* bout      = (const float*)d_in[10];
  float* out             = (float*)d_out;

  const int molecules = 32768 / NPM;  // 1024
  gcn_decoder_fused<<<molecules, 128, 0, stream>>>(
      h, node_mask, edge_mask, W0, b0, W1, b1, Wout, bout, out);
}